// KG_GCN_LSTMModule_42176578847420
// MI455X (gfx1250) — compile-verified
//
#include <hip/hip_runtime.h>
#include <math.h>

// ---------------------------------------------------------------------------
// GCN(128->64) -> ReLU -> GCN(64->64) -> 1-step LSTM on node 0 -> scalar.
// fp32 throughout (matches reference); dense GEMMs via V_WMMA_F32_16X16X4_F32.
// ---------------------------------------------------------------------------

typedef __attribute__((ext_vector_type(2))) float v2f;
typedef __attribute__((ext_vector_type(8))) float v8f;

#define GIN   128   // input feature dim
#define GH    64    // hidden dim (both GCN layers, LSTM hidden)

__device__ __forceinline__ v8f wmma_f32(v2f a, v2f b, v8f c) {
  // D = A(16x4,f32) * B(4x16,f32) + C(16x16,f32)
  return __builtin_amdgcn_wmma_f32_16x16x4_f32(
      /*neg_a=*/false, a, /*neg_b=*/false, b,
      /*c_mod=*/(short)0, c, /*reuse_a=*/false, /*reuse_b=*/false);
}

// ---------------- degree / normalization ------------------------------------
__global__ void k_deg_init(float* __restrict__ deg, int n) {
  int i = blockIdx.x * blockDim.x + threadIdx.x;
  if (i < n) deg[i] = 1.0f;              // self loop
}
__global__ void k_deg_count(const int* __restrict__ dst, float* __restrict__ deg, int e) {
  int i = blockIdx.x * blockDim.x + threadIdx.x;
  if (i < e) atomicAdd(&deg[dst[i]], 1.0f);
}
__global__ void k_rsqrt_inplace(float* __restrict__ d, int n) {
  int i = blockIdx.x * blockDim.x + threadIdx.x;
  if (i < n) d[i] = rsqrtf(d[i]);        // deg >= 1 always
}

// ---------------- GEMM1: out[N][64] = x^T[N][128] @ W[128][64] --------------
// x is stored feature-major: x[k*N + n].  One wave -> 16 rows x 64 cols.
__global__ void k_gemm_xw(const float* __restrict__ x, const float* __restrict__ W,
                          float* __restrict__ out, int N) {
  __shared__ __align__(16) float sW[GIN * GH];           // 32 KB
  const int t = threadIdx.x;
  for (int i = t; i < GIN * GH / 4; i += 256)
    ((float4*)sW)[i] = ((const float4*)W)[i];
  __syncthreads();

  const int lane = t & 31;
  const int rowTile = blockIdx.x * 8 + (t >> 5);
  if (rowTile * 16 >= N) return;                          // after barrier: safe
  const int row0 = rowTile * 16;
  const int ml = lane & 15;                               // M (and N) sub-index
  const int kh = (lane >> 4) * 2;                         // K offset 0 / 2

  v8f a0 = {}, a1 = {}, a2 = {}, a3 = {};
  for (int k0 = 0; k0 < GIN; k0 += 4) {
    const int ka = k0 + kh;
    v2f av;
    av.x = x[(size_t)ka * N + row0 + ml];
    av.y = x[(size_t)(ka + 1) * N + row0 + ml];
    const float* w0 = sW + ka * GH + ml;
    const float* w1 = sW + (ka + 1) * GH + ml;
    v2f b0; b0.x = w0[0];  b0.y = w1[0];
    v2f b1; b1.x = w0[16]; b1.y = w1[16];
    v2f b2; b2.x = w0[32]; b2.y = w1[32];
    v2f b3; b3.x = w0[48]; b3.y = w1[48];
    a0 = wmma_f32(av, b0, a0);
    a1 = wmma_f32(av, b1, a1);
    a2 = wmma_f32(av, b2, a2);
    a3 = wmma_f32(av, b3, a3);
  }
  const int mh = (lane >> 4) * 8;
#pragma unroll
  for (int r = 0; r < 8; ++r) {
    float* o = out + (size_t)(row0 + mh + r) * GH + ml;
    o[0] = a0[r]; o[16] = a1[r]; o[32] = a2[r]; o[48] = a3[r];
  }
}

// ---------------- GEMM2: out[N][64] = relu(A)[N][64] @ W[64][64] ------------
__global__ void k_gemm_relu_w(const float* __restrict__ A, const float* __restrict__ W,
                              float* __restrict__ out, int N) {
  __shared__ __align__(16) float sW[GH * GH];            // 16 KB
  const int t = threadIdx.x;
  for (int i = t; i < GH * GH / 4; i += 256)
    ((float4*)sW)[i] = ((const float4*)W)[i];
  __syncthreads();

  const int lane = t & 31;
  const int rowTile = blockIdx.x * 8 + (t >> 5);
  if (rowTile * 16 >= N) return;
  const int row0 = rowTile * 16;
  const int ml = lane & 15;
  const int kh = (lane >> 4) * 2;

  v8f a0 = {}, a1 = {}, a2 = {}, a3 = {};
  for (int k0 = 0; k0 < GH; k0 += 4) {
    const int ka = k0 + kh;
    const float* ap = A + (size_t)(row0 + ml) * GH + ka;
    v2f av;
    av.x = fmaxf(ap[0], 0.0f);                            // fused ReLU
    av.y = fmaxf(ap[1], 0.0f);
    const float* w0 = sW + ka * GH + ml;
    const float* w1 = sW + (ka + 1) * GH + ml;
    v2f b0; b0.x = w0[0];  b0.y = w1[0];
    v2f b1; b1.x = w0[16]; b1.y = w1[16];
    v2f b2; b2.x = w0[32]; b2.y = w1[32];
    v2f b3; b3.x = w0[48]; b3.y = w1[48];
    a0 = wmma_f32(av, b0, a0);
    a1 = wmma_f32(av, b1, a1);
    a2 = wmma_f32(av, b2, a2);
    a3 = wmma_f32(av, b3, a3);
  }
  const int mh = (lane >> 4) * 8;
#pragma unroll
  for (int r = 0; r < 8; ++r) {
    float* o = out + (size_t)(row0 + mh + r) * GH + ml;
    o[0] = a0[r]; o[16] = a1[r]; o[32] = a2[r]; o[48] = a3[r];
  }
}

// ---------------- agg[n][f] = t[n][f]*dinv[n]^2 + b[f]  (self loop + bias) --
__global__ void k_selfinit(const float* __restrict__ tin, const float* __restrict__ dinv,
                           const float* __restrict__ bias, float* __restrict__ agg, int N) {
  int idx = blockIdx.x * blockDim.x + threadIdx.x;
  if (idx >= N * GH) return;
  int n = idx >> 6, f = idx & 63;
  float di = dinv[n];
  agg[idx] = tin[idx] * di * di + bias[f];
}

// ---------------- edge scatter: agg[dst] += t[src] * dinv[src]*dinv[dst] ----
// 16 lanes per edge, float4 per lane -> contiguous 256B row gather, L2 atomics.
__global__ void k_scatter(const int* __restrict__ src, const int* __restrict__ dst,
                          const float* __restrict__ tin, const float* __restrict__ dinv,
                          float* __restrict__ agg, int E) {
  int gid = blockIdx.x * blockDim.x + threadIdx.x;
  int e = gid >> 4;
  if (e >= E) return;
  int fq = (gid & 15) * 4;
  int s = src[e], d = dst[e];
  float nrm = dinv[s] * dinv[d];
  const float4 v = *(const float4*)(tin + (size_t)s * GH + fq);
  float* o = agg + (size_t)d * GH + fq;
  atomicAdd(o + 0, v.x * nrm);
  atomicAdd(o + 1, v.y * nrm);
  atomicAdd(o + 2, v.z * nrm);
  atomicAdd(o + 3, v.w * nrm);
}

// ---------------- single-step LSTM + FC on node-0 embedding -----------------
__device__ __forceinline__ float sigmoidf(float x) { return 1.0f / (1.0f + expf(-x)); }

__global__ void k_lstm_head(const float* __restrict__ z,      // agg2 row 0 [64]
                            const float* __restrict__ w_ih,   // [256][64]
                            const float* __restrict__ b_ih, const float* __restrict__ b_hh,
                            const float* __restrict__ fc_w, const float* __restrict__ fc_b,
                            float* __restrict__ y) {
  __shared__ float gates[4 * GH];
  __shared__ float red[GH];
  const int t = threadIdx.x;                 // 256 threads = one gate each
  float acc = b_ih[t] + b_hh[t];             // h0 = c0 = 0 -> w_hh term vanishes
  const float* wr = w_ih + t * GH;
#pragma unroll 8
  for (int k = 0; k < GH; ++k) acc += z[k] * wr[k];
  gates[t] = acc;
  __syncthreads();
  if (t < GH) {
    float gi = gates[t];                     // i gate
    float gg = gates[2 * GH + t];            // g gate
    float go = gates[3 * GH + t];            // o gate
    float c  = sigmoidf(gi) * tanhf(gg);     // f*c0 = 0
    float hy = sigmoidf(go) * tanhf(c);
    red[t] = hy * fc_w[t];
  }
  __syncthreads();
  for (int s = 32; s > 0; s >>= 1) {
    if (t < s) red[t] += red[t + s];
    __syncthreads();
  }
  if (t == 0) y[0] = red[0] + fc_b[0];
}

// ---------------------------------------------------------------------------
extern "C" void kernel_launch(void* const* d_in, const int* in_sizes, int n_in,
                              void* d_out, int out_size, void* d_ws, size_t ws_size,
                              hipStream_t stream) {
  const float* x   = (const float*)d_in[0];
  const int*   ei  = (const int*)  d_in[1];
  const float* W1  = (const float*)d_in[2];
  const float* b1  = (const float*)d_in[3];
  const float* W2  = (const float*)d_in[4];
  const float* b2  = (const float*)d_in[5];
  const float* wih = (const float*)d_in[6];
  /* d_in[7] = w_hh: unused since h0 == 0 */
  const float* bih = (const float*)d_in[8];
  const float* bhh = (const float*)d_in[9];
  const float* fcw = (const float*)d_in[10];
  const float* fcb = (const float*)d_in[11];
  float* y = (float*)d_out;

  const int N = in_sizes[0] / GIN;   // 100000 (multiple of 16)
  const int E = in_sizes[1] / 2;     // 3200000
  const int* src = ei;
  const int* dst = ei + E;

  // workspace: bufT [N*64] | bufAgg [N*64] | dinv [N]   (~51.6 MB)
  char* ws = (char*)d_ws;
  float* bufT   = (float*)ws;
  float* bufAgg = (float*)(ws + (size_t)N * GH * sizeof(float));
  float* dinv   = (float*)(ws + (size_t)2 * N * GH * sizeof(float));

  const int BT = 256;
  dim3 blk(BT);
  int gN  = (N + BT - 1) / BT;
  int gE  = (E + BT - 1) / BT;
  int gNF = (N * GH + BT - 1) / BT;
  int gSC = (int)(((size_t)E * 16 + BT - 1) / BT);
  int gGm = ((N / 16) + 7) / 8;      // 8 waves (row tiles) per block

  // D^{-1/2} with self loops
  k_deg_init     <<<gN, blk, 0, stream>>>(dinv, N);
  k_deg_count    <<<gE, blk, 0, stream>>>(dst, dinv, E);
  k_rsqrt_inplace<<<gN, blk, 0, stream>>>(dinv, N);

  // layer 1: t1 = x^T @ W1 ; agg1 = norm-scatter(t1) + b1
  k_gemm_xw  <<<gGm, blk, 0, stream>>>(x, W1, bufT, N);
  k_selfinit <<<gNF, blk, 0, stream>>>(bufT, dinv, b1, bufAgg, N);
  k_scatter  <<<gSC, blk, 0, stream>>>(src, dst, bufT, dinv, bufAgg, E);

  // layer 2: t2 = relu(agg1) @ W2 ; agg2 = norm-scatter(t2) + b2
  k_gemm_relu_w<<<gGm, blk, 0, stream>>>(bufAgg, W2, bufT, N);
  k_selfinit   <<<gNF, blk, 0, stream>>>(bufT, dinv, b2, bufAgg, N);
  k_scatter    <<<gSC, blk, 0, stream>>>(src, dst, bufT, dinv, bufAgg, E);

  // LSTM + FC head on node 0
  k_lstm_head<<<1, blk, 0, stream>>>(bufAgg, wih, bih, bhh, fcw, fcb, y);
}